// VesselEncoder_36077725286933
// MI455X (gfx1250) — compile-verified
//
#include <hip/hip_runtime.h>
#include <hip/hip_bf16.h>
#include <math.h>

// ---------------------------------------------------------------------------
// GATv2 encoder (3 layers) for MI455X / gfx1250, wave32.
// Dense transforms use V_WMMA_F32_16X16X4_F32 (fp32 tensor path), software-
// pipelined so fragment loads for k-step n+1 overlap the WMMAs of step n.
// Edge softmax/aggregation: one wave32 per edge, coalesced float4 row access,
// fp32 global atomics for segment max / sum (L2-resident stats).
// ---------------------------------------------------------------------------

typedef __attribute__((ext_vector_type(2))) float v2f;
typedef __attribute__((ext_vector_type(8))) float v8f;

#define LRELU(x) fmaxf((x), 0.2f * (x))

__device__ __forceinline__ void atomicMaxFloat(float* addr, float val) {
    // monotone mapping trick; addr initialized to -inf and only touched here
    if (val >= 0.0f)
        atomicMax((int*)addr, __float_as_int(val));
    else
        atomicMin((unsigned int*)addr, __float_as_uint(val));
}

// ---------------------------------------------------------------------------
// Y[M,Nout] = X[M,K] @ W^T + bias ;  W is [Nout,K] row-major (PyTorch linear).
// One wave per 16 x (NT*16) slab; K stepped by 4 with v_wmma_f32_16x16x4_f32.
// A layout: lanes 0-15 -> M=lane, {K=k0,k0+1}; lanes 16-31 -> {K=k0+2,k0+3}.
// B layout mirrored (N=lane%16). D: VGPR j -> row j (lanes 0-15) / j+8.
// Fragment loads are pipelined one k-step ahead of the WMMAs.
// ---------------------------------------------------------------------------
template <int NT>
__global__ void linear_wmma(const float* __restrict__ X,
                            const float* __restrict__ W,
                            const float* __restrict__ bias,
                            float* __restrict__ Y,
                            int M, int K, int Nout) {
    int gw   = (blockIdx.x * blockDim.x + threadIdx.x) >> 5;
    int lane = threadIdx.x & 31;
    int ngroups = Nout / (NT * 16);
    int total   = (M / 16) * ngroups;
    if (gw >= total) return;

    int mt   = gw / ngroups;
    int ng   = gw % ngroups;
    int half = lane >> 4;       // which K-pair half
    int lm   = lane & 15;

    const float* xrow = X + (size_t)(mt * 16 + lm) * K + 2 * half;

    int cols[NT];
    const float* wrow[NT];
#pragma unroll
    for (int t = 0; t < NT; ++t) {
        cols[t] = ng * (NT * 16) + t * 16 + lm;
        wrow[t] = W + (size_t)cols[t] * K + 2 * half;
    }

    v8f acc[NT] = {};

    // --- prologue: load fragments for k0 = 0 ---
    v2f a_cur, b_cur[NT];
    a_cur.x = xrow[0];
    a_cur.y = xrow[1];
#pragma unroll
    for (int t = 0; t < NT; ++t) {
        b_cur[t].x = wrow[t][0];
        b_cur[t].y = wrow[t][1];
    }

    // --- pipelined main loop: load k0+4 while multiplying k0 ---
    for (int k0 = 0; k0 + 4 < K; k0 += 4) {
        v2f a_nxt, b_nxt[NT];
        a_nxt.x = xrow[k0 + 4];
        a_nxt.y = xrow[k0 + 5];
#pragma unroll
        for (int t = 0; t < NT; ++t) {
            b_nxt[t].x = wrow[t][k0 + 4];
            b_nxt[t].y = wrow[t][k0 + 5];
        }
#pragma unroll
        for (int t = 0; t < NT; ++t)
            acc[t] = __builtin_amdgcn_wmma_f32_16x16x4_f32(
                false, a_cur, false, b_cur[t], (short)0, acc[t], false, false);
        a_cur = a_nxt;
#pragma unroll
        for (int t = 0; t < NT; ++t) b_cur[t] = b_nxt[t];
    }

    // --- epilogue: last k-step ---
#pragma unroll
    for (int t = 0; t < NT; ++t)
        acc[t] = __builtin_amdgcn_wmma_f32_16x16x4_f32(
            false, a_cur, false, b_cur[t], (short)0, acc[t], false, false);

#pragma unroll
    for (int t = 0; t < NT; ++t) {
        float bv = bias[cols[t]];
#pragma unroll
        for (int j = 0; j < 8; ++j) {
            int m = mt * 16 + j + 8 * half;
            Y[(size_t)m * Nout + cols[t]] = acc[t][j] + bv;
        }
    }
}

// ---------------------------------------------------------------------------
__global__ void fillk(float* __restrict__ p, float v, int n) {
    int t = blockIdx.x * blockDim.x + threadIdx.x;
    if (t < n) p[t] = v;
}

// ---------------------------------------------------------------------------
// Layers 1/2: H=4 heads x C=64. One wave per edge; lane owns 8 channels.
// score[e,h] = sum_c att[h,c]*leaky(xl[src]+xr[dst]); atomic segment-max.
// ---------------------------------------------------------------------------
__global__ void edge_score4(const float* __restrict__ xl,
                            const float* __restrict__ xr,
                            const float* __restrict__ att,
                            const int* __restrict__ src,
                            const int* __restrict__ dst,
                            int E, int ET,
                            float* __restrict__ score,
                            float* __restrict__ smax) {
    int gw   = (blockIdx.x * blockDim.x + threadIdx.x) >> 5;
    int lane = threadIdx.x & 31;
    if (gw >= ET) return;
    int s, d;
    if (gw < E) { s = src[gw]; d = dst[gw]; } else { s = gw - E; d = s; }

    const float4* pl = (const float4*)(xl + (size_t)s * 256);
    const float4* pr = (const float4*)(xr + (size_t)d * 256);
    const float4* pa = (const float4*)att;
    float4 l0 = pl[lane * 2], l1 = pl[lane * 2 + 1];
    float4 r0 = pr[lane * 2], r1 = pr[lane * 2 + 1];
    float4 a0 = pa[lane * 2], a1 = pa[lane * 2 + 1];

    float p = 0.0f;
    p += a0.x * LRELU(l0.x + r0.x);
    p += a0.y * LRELU(l0.y + r0.y);
    p += a0.z * LRELU(l0.z + r0.z);
    p += a0.w * LRELU(l0.w + r0.w);
    p += a1.x * LRELU(l1.x + r1.x);
    p += a1.y * LRELU(l1.y + r1.y);
    p += a1.z * LRELU(l1.z + r1.z);
    p += a1.w * LRELU(l1.w + r1.w);

    // per-head reduce across 8-lane groups (head = lane/8)
    p += __shfl_xor(p, 1);
    p += __shfl_xor(p, 2);
    p += __shfl_xor(p, 4);

    int h = lane >> 3;
    if ((lane & 7) == 0) {
        score[(size_t)gw * 4 + h] = p;
        atomicMaxFloat(smax + (size_t)d * 4 + h, p);
    }
}

// exp(score - smax[dst]) in-place; accumulate denom. One thread per (edge,head).
__global__ void edge_exp(float* __restrict__ score,
                         const float* __restrict__ smax,
                         float* __restrict__ denom,
                         const int* __restrict__ dst,
                         int E, int ET, int H) {
    int t = blockIdx.x * blockDim.x + threadIdx.x;
    if (t >= ET * H) return;
    int e = t / H;
    int h = t - e * H;
    int d = (e < E) ? dst[e] : (e - E);
    float w = __expf(score[t] - smax[(size_t)d * H + h]);
    score[t] = w;
    atomicAdd(denom + (size_t)d * H + h, w);
}

// agg[dst] += (expw/denom[dst]) * xl[src]; one wave per edge, 8 ch/lane.
__global__ void edge_agg4(const float* __restrict__ xl,
                          const float* __restrict__ expw,
                          const float* __restrict__ denom,
                          const int* __restrict__ src,
                          const int* __restrict__ dst,
                          int E, int ET,
                          float* __restrict__ agg) {
    int gw   = (blockIdx.x * blockDim.x + threadIdx.x) >> 5;
    int lane = threadIdx.x & 31;
    if (gw >= ET) return;
    int s, d;
    if (gw < E) { s = src[gw]; d = dst[gw]; } else { s = gw - E; d = s; }

    int h   = lane >> 3;
    float w = expw[(size_t)gw * 4 + h] / (denom[(size_t)d * 4 + h] + 1e-16f);

    const float4* pl = (const float4*)(xl + (size_t)s * 256);
    float4 v0 = pl[lane * 2], v1 = pl[lane * 2 + 1];
    float* pa = agg + (size_t)d * 256 + lane * 8;
    atomicAdd(pa + 0, w * v0.x);
    atomicAdd(pa + 1, w * v0.y);
    atomicAdd(pa + 2, w * v0.z);
    atomicAdd(pa + 3, w * v0.w);
    atomicAdd(pa + 4, w * v1.x);
    atomicAdd(pa + 5, w * v1.y);
    atomicAdd(pa + 6, w * v1.z);
    atomicAdd(pa + 7, w * v1.w);
}

// h = elu(batchnorm(agg + conv_bias))
__global__ void node_post(const float* __restrict__ agg,
                          const float* __restrict__ bias,
                          const float* __restrict__ g,
                          const float* __restrict__ b,
                          const float* __restrict__ m,
                          const float* __restrict__ v,
                          float* __restrict__ out, int total, int C) {
    int t = blockIdx.x * blockDim.x + threadIdx.x;
    if (t >= total) return;
    int c = t & (C - 1);
    float x = agg[t] + bias[c];
    x = g[c] * (x - m[c]) * rsqrtf(v[c] + 1e-5f) + b[c];
    out[t] = (x > 0.0f) ? x : (__expf(x) - 1.0f);
}

// ---------------------------------------------------------------------------
// Layer 3: H=1, C=32. One wave per edge, one channel per lane.
// ---------------------------------------------------------------------------
__global__ void edge_score1(const float* __restrict__ xl,
                            const float* __restrict__ xr,
                            const float* __restrict__ att,
                            const int* __restrict__ src,
                            const int* __restrict__ dst,
                            int E, int ET,
                            float* __restrict__ score,
                            float* __restrict__ smax) {
    int gw   = (blockIdx.x * blockDim.x + threadIdx.x) >> 5;
    int lane = threadIdx.x & 31;
    if (gw >= ET) return;
    int s, d;
    if (gw < E) { s = src[gw]; d = dst[gw]; } else { s = gw - E; d = s; }

    float x = xl[(size_t)s * 32 + lane] + xr[(size_t)d * 32 + lane];
    float p = att[lane] * LRELU(x);
    p += __shfl_xor(p, 1);
    p += __shfl_xor(p, 2);
    p += __shfl_xor(p, 4);
    p += __shfl_xor(p, 8);
    p += __shfl_xor(p, 16);
    if (lane == 0) {
        score[gw] = p;
        atomicMaxFloat(smax + d, p);
    }
}

__global__ void edge_agg1(const float* __restrict__ xl,
                          const float* __restrict__ expw,
                          const float* __restrict__ denom,
                          const int* __restrict__ src,
                          const int* __restrict__ dst,
                          int E, int ET,
                          float* __restrict__ agg) {
    int gw   = (blockIdx.x * blockDim.x + threadIdx.x) >> 5;
    int lane = threadIdx.x & 31;
    if (gw >= ET) return;
    int s, d;
    if (gw < E) { s = src[gw]; d = dst[gw]; } else { s = gw - E; d = s; }
    float w = expw[gw] / (denom[d] + 1e-16f);
    atomicAdd(agg + (size_t)d * 32 + lane, w * xl[(size_t)s * 32 + lane]);
}

__global__ void add_bias(const float* __restrict__ agg,
                         const float* __restrict__ bias,
                         float* __restrict__ out, int total, int C) {
    int t = blockIdx.x * blockDim.x + threadIdx.x;
    if (t >= total) return;
    out[t] = agg[t] + bias[t & (C - 1)];
}

// ---------------------------------------------------------------------------
extern "C" void kernel_launch(void* const* d_in, const int* in_sizes, int n_in,
                              void* d_out, int out_size, void* d_ws, size_t ws_size,
                              hipStream_t stream) {
    const int N = 20000, E = 320000, ET = 340000;
    const float NEG_INF = -__builtin_huge_valf();

    const float* x    = (const float*)d_in[0];
    const int*   ei   = (const int*)d_in[1];
    const int*   src  = ei;
    const int*   dst  = ei + E;

    const float* Wl1 = (const float*)d_in[2];  const float* bl1 = (const float*)d_in[3];
    const float* Wr1 = (const float*)d_in[4];  const float* br1 = (const float*)d_in[5];
    const float* att1 = (const float*)d_in[6]; const float* bias1 = (const float*)d_in[7];
    const float* g1 = (const float*)d_in[8];   const float* b1 = (const float*)d_in[9];
    const float* m1 = (const float*)d_in[10];  const float* v1 = (const float*)d_in[11];

    const float* Wl2 = (const float*)d_in[12]; const float* bl2 = (const float*)d_in[13];
    const float* Wr2 = (const float*)d_in[14]; const float* br2 = (const float*)d_in[15];
    const float* att2 = (const float*)d_in[16];const float* bias2 = (const float*)d_in[17];
    const float* g2 = (const float*)d_in[18];  const float* b2 = (const float*)d_in[19];
    const float* m2 = (const float*)d_in[20];  const float* v2 = (const float*)d_in[21];

    const float* Wl3 = (const float*)d_in[22]; const float* bl3 = (const float*)d_in[23];
    const float* Wr3 = (const float*)d_in[24]; const float* br3 = (const float*)d_in[25];
    const float* att3 = (const float*)d_in[26];const float* bias3 = (const float*)d_in[27];

    float* out = (float*)d_out;

    // ------- workspace carve (floats) -------
    float* ws = (float*)d_ws;
    const size_t nf = (size_t)N * 256;
    float* hbuf = ws;                       // N*256
    float* xl   = ws + nf;                  // N*256 (layer3: N*32 prefix)
    float* xr   = ws + 2 * nf;              // N*256
    float* agg  = ws + 3 * nf;              // N*256
    float* sc   = ws + 4 * nf;              // ET*4 (score -> expw in place)
    float* smax = sc + (size_t)ET * 4;      // N*4
    float* den  = smax + (size_t)N * 4;     // N*4

    dim3 blk(256);
    const int lin256 = ((N / 16) * 4 * 32 + 255) / 256;   // NT=4 slabs over 256 cols
    const int lin32  = ((N / 16) * 1 * 32 + 255) / 256;   // NT=2 slab over 32 cols
    const int ewave  = (ET + 7) / 8;                      // 8 waves / block
    const int eH4    = (ET * 4 + 255) / 256;
    const int eH1    = (ET + 255) / 256;
    const int npB    = (N * 256 / 256);
    const int fNH4   = (N * 4 + 255) / 256;
    const int fN     = (N + 255) / 256;
    const int fN32   = (N * 32 + 255) / 256;

    // ================= layer 1 (4 -> 4x64, concat) =================
    linear_wmma<4><<<lin256, blk, 0, stream>>>(x, Wl1, bl1, xl, N, 4, 256);
    linear_wmma<4><<<lin256, blk, 0, stream>>>(x, Wr1, br1, xr, N, 4, 256);
    fillk<<<fNH4, blk, 0, stream>>>(smax, NEG_INF, N * 4);
    fillk<<<fNH4, blk, 0, stream>>>(den, 0.0f, N * 4);
    fillk<<<npB,  blk, 0, stream>>>(agg, 0.0f, N * 256);
    edge_score4<<<ewave, blk, 0, stream>>>(xl, xr, att1, src, dst, E, ET, sc, smax);
    edge_exp<<<eH4, blk, 0, stream>>>(sc, smax, den, dst, E, ET, 4);
    edge_agg4<<<ewave, blk, 0, stream>>>(xl, sc, den, src, dst, E, ET, agg);
    node_post<<<npB, blk, 0, stream>>>(agg, bias1, g1, b1, m1, v1, hbuf, N * 256, 256);

    // ================= layer 2 (256 -> 4x64, concat) =================
    linear_wmma<4><<<lin256, blk, 0, stream>>>(hbuf, Wl2, bl2, xl, N, 256, 256);
    linear_wmma<4><<<lin256, blk, 0, stream>>>(hbuf, Wr2, br2, xr, N, 256, 256);
    fillk<<<fNH4, blk, 0, stream>>>(smax, NEG_INF, N * 4);
    fillk<<<fNH4, blk, 0, stream>>>(den, 0.0f, N * 4);
    fillk<<<npB,  blk, 0, stream>>>(agg, 0.0f, N * 256);
    edge_score4<<<ewave, blk, 0, stream>>>(xl, xr, att2, src, dst, E, ET, sc, smax);
    edge_exp<<<eH4, blk, 0, stream>>>(sc, smax, den, dst, E, ET, 4);
    edge_agg4<<<ewave, blk, 0, stream>>>(xl, sc, den, src, dst, E, ET, agg);
    node_post<<<npB, blk, 0, stream>>>(agg, bias2, g2, b2, m2, v2, hbuf, N * 256, 256);

    // ================= layer 3 (256 -> 32, 1 head) =================
    linear_wmma<2><<<lin32, blk, 0, stream>>>(hbuf, Wl3, bl3, xl, N, 256, 32);
    linear_wmma<2><<<lin32, blk, 0, stream>>>(hbuf, Wr3, br3, xr, N, 256, 32);
    fillk<<<fN,   blk, 0, stream>>>(smax, NEG_INF, N);
    fillk<<<fN,   blk, 0, stream>>>(den, 0.0f, N);
    fillk<<<fN32, blk, 0, stream>>>(agg, 0.0f, N * 32);
    edge_score1<<<ewave, blk, 0, stream>>>(xl, xr, att3, src, dst, E, ET, sc, smax);
    edge_exp<<<eH1, blk, 0, stream>>>(sc, smax, den, dst, E, ET, 1);
    edge_agg1<<<ewave, blk, 0, stream>>>(xl, sc, den, src, dst, E, ET, agg);
    add_bias<<<fN32, blk, 0, stream>>>(agg, bias3, out, N * 32, 32);
}